// RoiPoolingConv_20985210208519
// MI455X (gfx1250) — compile-verified
//
#include <hip/hip_runtime.h>

// ROI-Align bilinear pooling, MI455X (gfx1250).
// Memory-bound: ~84 MB HBM traffic -> ~3.6us floor @ 23.3 TB/s.
// Data path: gfx1250 async data mover (global_load_async_to_lds_b128 /
// s_wait_asynccnt) stages the 4 bilinear source pixels (4 x 2KB, NHWC
// contiguous) into LDS; compute is a trivial 2-level lerp; output uses
// non-temporal stores so the 32MB feature map stays L2-resident.
// Grid is (7,7,R) so px/py/roi come straight from blockIdx (no int division).

#define POOL 7
#define HF   128
#define WF   128
#define CCH  512          // channels
#define TPB  128          // threads/block; thread t owns channels 4t..4t+3

typedef float v4f __attribute__((ext_vector_type(4)));

__device__ __forceinline__ void async_ld16(unsigned lds_off, const float* g) {
    // per-lane 16B global -> LDS, tracked by ASYNCcnt
    asm volatile("global_load_async_to_lds_b128 %0, %1, off"
                 :: "v"(lds_off), "v"((unsigned long long)g)
                 : "memory");
}

__global__ __launch_bounds__(TPB) void roi_align_kernel(
    const float* __restrict__ img, const float* __restrict__ rois,
    float* __restrict__ out)
{
    __shared__ v4f tile[4 * (CCH / 4)];   // 4 pixels x 512 floats = 8 KB

    const int t  = threadIdx.x;           // channel-group id (0..127)
    const int px = blockIdx.x;            // 0..6
    const int py = blockIdx.y;            // 0..6
    const int r  = blockIdx.z;            // 0..R-1

    // ROI box -> integer corners (truncating cast, values are non-negative)
    const v4f roi = *(const v4f*)(rois + 4 * r);
    const int r1 = (int)(roi.x * (float)HF);
    const int c1 = (int)(roi.y * (float)WF);
    const int r2 = (int)(roi.z * (float)HF);
    const int c2 = (int)(roi.w * (float)WF);
    const int hs = max(r2 - r1, 1);
    const int ws = max(c2 - c1, 1);

    // sample position + bilinear weights (matches reference op order)
    const float ry  = (float)py * ((float)hs / (float)POOL);
    const float rx  = (float)px * ((float)ws / (float)POOL);
    const float ry0 = floorf(ry);
    const float rx0 = floorf(rx);
    const float fy  = ry - ry0;
    const float fx  = rx - rx0;
    const int y0 = min(max((int)ry0, 0), hs - 1);
    const int y1 = min(y0 + 1, hs - 1);
    const int x0 = min(max((int)rx0, 0), ws - 1);
    const int x1 = min(x0 + 1, ws - 1);
    const int Y0 = r1 + y0, Y1 = r1 + y1;
    const int X0 = c1 + x0, X1 = c1 + x1;

    const float* pa = img + (size_t)(Y0 * WF + X0) * CCH;
    const float* pb = img + (size_t)(Y0 * WF + X1) * CCH;
    const float* pc = img + (size_t)(Y1 * WF + X0) * CCH;
    const float* pd = img + (size_t)(Y1 * WF + X1) * CCH;

    // LDS byte offset = low 32 bits of the generic (aperture|offset) address
    const unsigned lds_base = (unsigned)(unsigned long long)(void*)tile;
    const unsigned lane16   = (unsigned)t * 16u;

    async_ld16(lds_base + 0u * 2048u + lane16, pa + 4 * t);
    async_ld16(lds_base + 1u * 2048u + lane16, pb + 4 * t);
    async_ld16(lds_base + 2u * 2048u + lane16, pc + 4 * t);
    async_ld16(lds_base + 3u * 2048u + lane16, pd + 4 * t);
    asm volatile("s_wait_asynccnt 0x0" ::: "memory");
    // Each lane reads back exactly the bytes its own async load wrote,
    // so the per-wave ASYNCcnt wait is a sufficient ordering point.

    const v4f a = tile[0 * (CCH / 4) + t];
    const v4f b = tile[1 * (CCH / 4) + t];
    const v4f c = tile[2 * (CCH / 4) + t];
    const v4f d = tile[3 * (CCH / 4) + t];

    const v4f top = a + (b - a) * fx;
    const v4f bot = c + (d - c) * fx;
    const v4f o   = top + (bot - top) * fy;

    // out[((r*7+py)*7+px)*512 + 4t ...], write-once -> non-temporal store
    const size_t bid = ((size_t)r * POOL + py) * POOL + px;
    __builtin_nontemporal_store(o, (v4f*)out + bid * (CCH / 4) + t);
}

extern "C" void kernel_launch(void* const* d_in, const int* in_sizes, int n_in,
                              void* d_out, int out_size, void* d_ws, size_t ws_size,
                              hipStream_t stream) {
    const float* img  = (const float*)d_in[0];   // (1,128,128,512) f32
    const float* rois = (const float*)d_in[1];   // (1,R,4) f32
    // d_in[2] = pool_size scalar (==7), baked in as POOL
    float* out = (float*)d_out;                  // (1,R,7,7,512) f32

    const int R = in_sizes[1] / 4;
    hipLaunchKernelGGL(roi_align_kernel, dim3(POOL, POOL, R), dim3(TPB), 0,
                       stream, img, rois, out);
}